// Agito_59871844106371
// MI455X (gfx1250) — compile-verified
//
#include <hip/hip_runtime.h>

// Problem constants from the reference
#define BN 64
#define SN 32768
#define DN 4
#define VN 155
#define NTOK (BN * SN)          // 2,097,152 tokens
#define WAVES_PER_BLOCK 8
#define TOKENS_PER_TILE 16

typedef __attribute__((ext_vector_type(2))) float v2f;
typedef __attribute__((ext_vector_type(8))) float v8f;

// Fast sigmoid: v_exp_f32 + v_rcp_f32 (~1 ulp), avoids the IEEE div-fixup
// sequence (div_scale/rcp/fmas/fixup ~10 VALU ops) the precise '/' emits.
__device__ __forceinline__ float sigmoidf_(float x) {
    return __builtin_amdgcn_rcpf(1.0f + __expf(-x));
}

// B1 (for current-token WMMA): col 0-3 = WB^T, col 8 = WB^T w1,
// col 10 = w1, col 11 = w2, col 12 = Wc, col 13 = WB^T Wc, rest 0.
__device__ __forceinline__ float b1_col(int n, int d,
                                        const float* __restrict__ WB,
                                        const float* __restrict__ w1,
                                        const float* __restrict__ w2,
                                        const float* __restrict__ Wc) {
    if (n < 4) return WB[n * 4 + d];
    if (n == 8) {
        float s = 0.0f;
        #pragma unroll
        for (int e = 0; e < 4; ++e) s += WB[e * 4 + d] * w1[e];
        return s;
    }
    if (n == 10) return w1[d];
    if (n == 11) return w2[d];
    if (n == 12) return Wc[d];
    if (n == 13) {
        float s = 0.0f;
        #pragma unroll
        for (int e = 0; e < 4; ++e) s += WB[e * 4 + d] * Wc[e];
        return s;
    }
    return 0.0f;
}

// B2 (for next-token WMMA): col 0-3 = WBf^T, col 8 = WBf^T w2,
// col 13 = WBf^T Wc, rest 0.
__device__ __forceinline__ float b2_col(int n, int d,
                                        const float* __restrict__ WBf,
                                        const float* __restrict__ w2,
                                        const float* __restrict__ Wc) {
    if (n < 4) return WBf[n * 4 + d];
    if (n == 8) {
        float s = 0.0f;
        #pragma unroll
        for (int e = 0; e < 4; ++e) s += WBf[e * 4 + d] * w2[e];
        return s;
    }
    if (n == 13) {
        float s = 0.0f;
        #pragma unroll
        for (int e = 0; e < 4; ++e) s += WBf[e * 4 + d] * Wc[e];
        return s;
    }
    return 0.0f;
}

__global__ __launch_bounds__(256) void gated_fuse_wmma(
    const int* __restrict__ idx, const float* __restrict__ emb,
    const float* __restrict__ WB, const float* __restrict__ WBf,
    const float* __restrict__ w1, const float* __restrict__ w2,
    const float* __restrict__ bsc, const float* __restrict__ Wc,
    float* __restrict__ out_y, float* __restrict__ out_g,
    float* __restrict__ out_f)
{
    __shared__ float s_emb[VN * 4];
    __shared__ float s_y[WAVES_PER_BLOCK][16][4];
    __shared__ float s_g[WAVES_PER_BLOCK][16];
    __shared__ float s_fo[WAVES_PER_BLOCK][16];

    // Stage the tiny embedding table (2.4 KB) once per block.
    for (int i = threadIdx.x; i < VN * 4; i += 256) s_emb[i] = emb[i];
    __syncthreads();

    const int lane = threadIdx.x & 31;
    const int wave = threadIdx.x >> 5;
    const int tile = blockIdx.x * WAVES_PER_BLOCK + wave;
    const int t0   = tile * TOKENS_PER_TILE;   // global token base of this tile
    const int ts0  = t0 & (SN - 1);            // position within its sequence

    // ---- A matrices: 16x4 f32, lane L<16 holds row L (K=0,1), L>=16 row L-16 (K=2,3)
    const int row = lane & 15;
    const int kp  = lane >> 4;        // 0 -> components 0,1 ; 1 -> components 2,3
    const int ta  = t0 + row;
    int ta1 = ta + 1; if (ta1 >= NTOK) ta1 = NTOK - 1;   // clamp (overridden at seq end)
    const int ea  = idx[ta]  * 4;
    const int ea1 = idx[ta1] * 4;
    v2f A, An;
    A.x  = s_emb[ea  + 2 * kp]; A.y  = s_emb[ea  + 2 * kp + 1];
    An.x = s_emb[ea1 + 2 * kp]; An.y = s_emb[ea1 + 2 * kp + 1];

    // ---- B matrices: 4x16 f32, lane L holds (K=2h, N=n) in .x, (K=2h+1, N=n) in .y
    const int n = lane & 15;
    const int h = lane >> 4;
    v2f B1v, B2v;
    B1v.x = b1_col(n, 2 * h,     WB,  w1, w2, Wc);
    B1v.y = b1_col(n, 2 * h + 1, WB,  w1, w2, Wc);
    B2v.x = b2_col(n, 2 * h,     WBf, w2, Wc);
    B2v.y = b2_col(n, 2 * h + 1, WBf, w2, Wc);

    // ---- One WMMA per operand set: D[m][n] = sum_d u[t0+m][d] * Bcol_n[d]
    v8f C = {};
    v8f Dm = __builtin_amdgcn_wmma_f32_16x16x4_f32(false, A,  false, B1v,
                                                   (short)0, C, false, false);
    v8f Dx = __builtin_amdgcn_wmma_f32_16x16x4_f32(false, An, false, B2v,
                                                   (short)0, C, false, false);

    const float b0    = bsc[0];
    const int   sbase = h << 4;       // shuffle source base for this half-wave

    #pragma unroll
    for (int r = 0; r < 8; ++r) {
        const int   m   = r + 8 * h;          // token slot in tile for this half
        const int   ts  = ts0 + m;            // position within the sequence
        const bool  isF = (ts == 0);
        const bool  isL = (ts == SN - 1);

        // u for this token in D-layout (lane n<4 holds component n); reuse the
        // idx value already sitting in lane m's 'ea' register.
        const int   eix = __shfl(ea, m);
        const float u_n = (n < 4) ? s_emb[eix + n] : 0.0f;

        const float Dr  = Dm[r];
        const float Dxr = Dx[r];

        // Broadcast the dot-product columns within each 16-lane half.
        const float d8  = __shfl(Dr,  sbase + 8);    // x_f . w1   (interior)
        const float d10 = __shfl(Dr,  sbase + 10);   // u . w1     (t == 0)
        const float d11 = __shfl(Dr,  sbase + 11);   // u . w2     (t == S-1)
        const float d12 = __shfl(Dr,  sbase + 12);   // u . Wc
        const float d13 = __shfl(Dr,  sbase + 13);   // x_f . Wc   (interior)
        const float e8  = __shfl(Dxr, sbase + 8);    // x_b . w2   (interior)
        const float e13 = __shfl(Dxr, sbase + 13);   // x_b . Wc   (interior)

        const float sF = isF ? d10 : d8;
        const float sB = isL ? d11 : e8;
        const float g  = sigmoidf_(sF + sB + b0);

        const float cF = isF ? d12 : d13;
        const float cB = isL ? d12 : e13;
        const float fo = sigmoidf_(g * (cF + cB) + (1.0f - g) * d12);

        const float xf = isF ? u_n : Dr;
        const float xb = isL ? u_n : Dxr;
        const float y  = g * (xf + xb) + (1.0f - g) * u_n;

        if (n < 4)  s_y[wave][m][n] = y;
        if (n == 0) { s_g[wave][m] = g; s_fo[wave][m] = fo; }
    }
    __syncthreads();

    // Coalesced write-out: lane m (< 16) owns token t0+m; y goes out as B128.
    if (lane < 16) {
        const int tok = t0 + lane;
        float4 yv = make_float4(s_y[wave][lane][0], s_y[wave][lane][1],
                                s_y[wave][lane][2], s_y[wave][lane][3]);
        *reinterpret_cast<float4*>(out_y + (size_t)tok * 4) = yv;
        out_g[tok] = s_g[wave][lane];
        out_f[tok] = s_fo[wave][lane];
    }
}

extern "C" void kernel_launch(void* const* d_in, const int* in_sizes, int n_in,
                              void* d_out, int out_size, void* d_ws, size_t ws_size,
                              hipStream_t stream) {
    (void)in_sizes; (void)n_in; (void)out_size; (void)d_ws; (void)ws_size;
    // setup_inputs order: idx, emb, WA, WB, WAf, WBf, w1, w2, b, Wc
    const int*   idx = (const int*)  d_in[0];
    const float* emb = (const float*)d_in[1];
    // d_in[2] = WA, d_in[4] = WAf : multiply zeros in the reference, unused.
    const float* WB  = (const float*)d_in[3];
    const float* WBf = (const float*)d_in[5];
    const float* w1  = (const float*)d_in[6];
    const float* w2  = (const float*)d_in[7];
    const float* b   = (const float*)d_in[8];
    const float* Wc  = (const float*)d_in[9];

    float* out   = (float*)d_out;
    float* out_y = out;                              // [B,S,D]
    float* out_g = out + (size_t)NTOK * DN;          // [B,S,1]
    float* out_f = out + (size_t)NTOK * (DN + 1);    // [B,S]

    const int tiles  = NTOK / TOKENS_PER_TILE;       // 131072
    const int blocks = tiles / WAVES_PER_BLOCK;      // 16384
    gated_fuse_wmma<<<blocks, 256, 0, stream>>>(idx, emb, WB, WBf, w1, w2, b, Wc,
                                                out_y, out_g, out_f);
}